// ParallelBackproj2D_29850022707684
// MI455X (gfx1250) — compile-verified
//
#include <hip/hip_runtime.h>
#include <math.h>

#define NX    512
#define NY    512
#define ND    736
#define NA    720
#define NB    8
#define TILE  16
#define WIN   32           // staged detector window (floats) per batch
#define CHUNK 4            // angles staged per buffer
#define DETC  367.5f       // (ND-1)/2

typedef __attribute__((ext_vector_type(2))) float v2f;
typedef __attribute__((ext_vector_type(4))) float v4f;

// LDS byte offset of a generic pointer to __shared__ data:
// flat LDS addresses carry the LDS offset in the low 32 bits.
__device__ __forceinline__ unsigned lds_off(const void* p) {
    return (unsigned)(unsigned long long)(uintptr_t)p;
}

// paired gather: loads window[li + O0] and window[li + O1] in one DS op
#define GATHER2(P, O0, O1)                                                \
    asm volatile("ds_load_2addr_b32 %0, %1 offset0:" #O0 " offset1:" #O1  \
                 : "=v"(P) : "v"(waddr) : "memory")

__global__ __launch_bounds__(256)
void ParallelBackproj2D_29850022707684_kernel(const float* __restrict__ sino,
                                              float* __restrict__ out)
{
    __shared__ v4f s_ang[NA];                            // {cos, sin, ibase(bits), 0}
    __shared__ __align__(16) float s_win[2][CHUNK][NB][WIN];

    const int t  = threadIdx.x;
    const int bx = blockIdx.x;
    const int by = blockIdx.y;

    // tile center (block-uniform)
    const float xc = (float)(bx * TILE) + 7.5f - 255.5f;
    const float yc = (float)(by * TILE) + 7.5f - 255.5f;

    // ---- per-block angle table: cos, sin, aligned window base ----
    for (int a = t; a < NA; a += 256) {
        float th = (float)a * (float)(M_PI / (double)NA);
        float sv, cv;
        __sincosf(th, &sv, &cv);
        float g  = fabsf(cv) + fabsf(sv);
        float uc = fmaf(xc, cv, fmaf(yc, sv, DETC));
        int ib = ((int)floorf(fmaf(-7.5f, g, uc))) & ~3;   // 16B-aligned window start
        ib = ib < 0 ? 0 : (ib > ND - WIN ? ND - WIN : ib);
        v4f e; e.x = cv; e.y = sv; e.z = __int_as_float(ib); e.w = 0.0f;
        s_ang[a] = e;
    }

    // voxel handled by this thread (consecutive t -> consecutive y: coalesced stores)
    const int lx = t >> 4;
    const int ly = t & 15;
    const int x  = bx * TILE + lx;
    const int y  = by * TILE + ly;
    const float xf = (float)x - 255.5f;
    const float yf = (float)y - 255.5f;

    float acc[NB];
#pragma unroll
    for (int b = 0; b < NB; ++b) acc[b] = 0.0f;

    __syncthreads();   // angle table ready

    // prefetch mapping: all 256 threads move one float4 each per chunk
    const int ak = t >> 6;          // angle slot within chunk (0..3)
    const int pb = (t >> 3) & 7;    // batch (0..7)
    const int pj = (t & 7) * 4;     // float4 slot within window

    // issue async prefetch for one chunk of 4 angles
    auto issue_prefetch = [&](int chunk, int buf) {
        int a1 = chunk * CHUNK + ak;
        v4f ang = s_ang[a1];
        int ib  = __float_as_int(ang.z);
        unsigned goff = (unsigned)((pb * NA + a1) * ND + ib + pj) * 4u;
        unsigned loff = lds_off(&s_win[buf][ak][pb][pj]);
        asm volatile("global_load_async_to_lds_b128 %0, %1, %2"
                     :: "v"(loff), "v"(goff), "s"(sino) : "memory");
    };

    issue_prefetch(0, 0);

    for (int ch = 0; ch < NA / CHUNK; ++ch) {
        const int buf = ch & 1;

        // windows[buf] landed; all waves done reading windows[buf^1]
        asm volatile("s_wait_asynccnt 0x0" ::: "memory");
        __syncthreads();

        // overlap: stage next chunk while computing this one
        if (ch + 1 < NA / CHUNK) issue_prefetch(ch + 1, buf ^ 1);

#pragma unroll
        for (int k = 0; k < CHUNK; ++k) {
            v4f ang = s_ang[ch * CHUNK + k];
            float u  = fmaf(xf, ang.x, fmaf(yf, ang.y, DETC));
            float fi = floorf(u);
            float w  = u - fi;
            float w0 = 1.0f - w;
            int   li = (int)fi - __float_as_int(ang.z);     // 0..30 by construction
            unsigned waddr = lds_off(&s_win[buf][k][0][li]);

            v2f p0, p1, p2, p3, p4, p5, p6, p7;
            GATHER2(p0,   0,   1);
            GATHER2(p1,  32,  33);
            GATHER2(p2,  64,  65);
            GATHER2(p3,  96,  97);
            GATHER2(p4, 128, 129);
            GATHER2(p5, 160, 161);
            GATHER2(p6, 192, 193);
            GATHER2(p7, 224, 225);
            // tie all results to the wait so no consumer is hoisted above it
            asm volatile("s_wait_dscnt 0x0"
                         : "+v"(p0), "+v"(p1), "+v"(p2), "+v"(p3),
                           "+v"(p4), "+v"(p5), "+v"(p6), "+v"(p7)
                         :: "memory");

            acc[0] = fmaf(p0.x, w0, fmaf(p0.y, w, acc[0]));
            acc[1] = fmaf(p1.x, w0, fmaf(p1.y, w, acc[1]));
            acc[2] = fmaf(p2.x, w0, fmaf(p2.y, w, acc[2]));
            acc[3] = fmaf(p3.x, w0, fmaf(p3.y, w, acc[3]));
            acc[4] = fmaf(p4.x, w0, fmaf(p4.y, w, acc[4]));
            acc[5] = fmaf(p5.x, w0, fmaf(p5.y, w, acc[5]));
            acc[6] = fmaf(p6.x, w0, fmaf(p6.y, w, acc[6]));
            acc[7] = fmaf(p7.x, w0, fmaf(p7.y, w, acc[7]));
        }
    }

    const float scale = 0.0043633231299858238f;   // pi / 720
#pragma unroll
    for (int b = 0; b < NB; ++b) {
        out[((b * NX + x) * NY) + y] = acc[b] * scale;
    }
}

extern "C" void kernel_launch(void* const* d_in, const int* in_sizes, int n_in,
                              void* d_out, int out_size, void* d_ws, size_t ws_size,
                              hipStream_t stream) {
    (void)in_sizes; (void)n_in; (void)d_ws; (void)ws_size; (void)out_size;
    const float* sino = (const float*)d_in[0];   // [8,720,736,1] f32
    float* out = (float*)d_out;                  // [8,512,512,1] f32
    dim3 grid(NX / TILE, NY / TILE);
    dim3 block(256);
    ParallelBackproj2D_29850022707684_kernel<<<grid, block, 0, stream>>>(sino, out);
}